// MAF_83399674954371
// MI455X (gfx1250) — compile-verified
//
#include <hip/hip_runtime.h>
#include <hip/hip_bf16.h>
#include <math.h>

typedef __attribute__((ext_vector_type(2))) float v2f;
typedef __attribute__((ext_vector_type(8))) float v8f;

// ---- problem constants ----
#define BATCH 65536
#define DD 16
#define HH 64
#define NM 3

// ---- LDS layout (floats) ----
// Weights stored K-pair interleaved: wp[k/2][n][2] = (w[k][n], w[k+1][n])
// Row stride == 32 (mod 64) dwords -> lane halves (k-pair vs k-pair+1) hit
// disjoint bank halves -> conflict-free ds_load_b64 B fragments.
#define W01_S 160             // pair-row stride for N=64 weights (>=128, ==32 mod 64)
#define W2_S  32              // pair-row stride for N=16 weights  (>=32, ==32 mod 64)
#define OFF_W0P 0             // 8 pair-rows  * 160 = 1280
#define OFF_W1P 1280          // 32 pair-rows * 160 = 5120
#define OFF_W2P 6400          // 32 pair-rows * 32  = 1024
#define NET_SZ  7424
#define WGT_SZ  (2*NET_SZ)    // 14848

#define XW_S    18            // even -> b64-aligned A fragments
#define ACT_S   66            // even -> b64-aligned A fragments
#define XW_SZ   (16*XW_S)     // 288
#define ACT_SZ  (16*ACT_S)    // 1056
#define WAVE_SCR (XW_SZ + 2*ACT_SZ)  // 2400
#define WAVES_PER_BLOCK 8
#define LDS_FLOATS (WGT_SZ + WAVES_PER_BLOCK*WAVE_SCR)  // 34048
#define LDS_BYTES  (LDS_FLOATS*4)                       // 136192

#define LOG2E 1.4426950408889634f

// branch-free fast tanh/sigmoid on hardware transcendentals (v_exp_f32, v_rcp_f32)
__device__ __forceinline__ float fast_tanh(float x) {
    float e = __builtin_amdgcn_exp2f(x * (2.0f * LOG2E));   // exp(2x)
    return 1.0f - 2.0f * __builtin_amdgcn_rcpf(e + 1.0f);
}
__device__ __forceinline__ float fast_sigmoid(float a) {
    return __builtin_amdgcn_rcpf(1.0f + __builtin_amdgcn_exp2f(a * -LOG2E));
}

__global__ void __launch_bounds__(256, 1)
maf_wmma_kernel(const float* __restrict__ mu_M0, const float* __restrict__ mu_M1,
                const float* __restrict__ mu_M2, const float* __restrict__ mu_W0,
                const float* __restrict__ mu_b0, const float* __restrict__ mu_W1,
                const float* __restrict__ mu_b1, const float* __restrict__ mu_W2,
                const float* __restrict__ mu_b2,
                const float* __restrict__ al_M0, const float* __restrict__ al_M1,
                const float* __restrict__ al_M2, const float* __restrict__ al_W0,
                const float* __restrict__ al_b0, const float* __restrict__ al_W1,
                const float* __restrict__ al_b1, const float* __restrict__ al_W2,
                const float* __restrict__ al_b2,
                const int* __restrict__ perms,
                const float* __restrict__ u, float* __restrict__ out)
{
    extern __shared__ float lds[];
    const int tid  = threadIdx.x;
    const int lane = tid & 31;
    const int wv   = tid >> 5;
    const int hl   = lane >> 4;    // lane half (0/1) -> selects K pair
    const int l15  = lane & 15;

    float* wgt  = lds;                               // both nets' masked weights (shared)
    float* xw   = lds + WGT_SZ + wv * WAVE_SCR;      // [16][18] x tile   (wave-private)
    float* actA = xw + XW_SZ;                        // [16][66]          (wave-private)
    float* actB = actA + ACT_SZ;                     // [16][66]          (wave-private)

    const int sampleBase = (blockIdx.x * WAVES_PER_BLOCK + wv) * 16;

    // load raw x tile (16 samples x 16 dims)
    for (int idx = lane; idx < 256; idx += 32) {
        int m = idx >> 4, d = idx & 15;
        xw[m * XW_S + d] = u[(size_t)(sampleBase + m) * DD + d];
    }

    for (int flow = 0; flow < NM; ++flow) {
        __syncthreads();  // previous flow done with shared weights; xw writes visible

        // ---- build masked, transposed, K-pair-interleaved weights (cooperative) ----
        for (int e = tid; e < HH * DD; e += 256) {       // W0: K=16, N=64
            int k = e & 15, n = e >> 4;
            int src = flow * HH * DD + n * DD + k;       // M0[n][k]*W0[n][k]
            int dst = OFF_W0P + (k >> 1) * W01_S + n * 2 + (k & 1);
            wgt[dst]          = mu_M0[src] * mu_W0[src];
            wgt[NET_SZ + dst] = al_M0[src] * al_W0[src];
        }
        for (int e = tid; e < HH * HH; e += 256) {       // W1: K=64, N=64
            int k = e & 63, n = e >> 6;
            int src = flow * HH * HH + n * HH + k;       // M1[n][k]*W1[n][k]
            int dst = OFF_W1P + (k >> 1) * W01_S + n * 2 + (k & 1);
            wgt[dst]          = mu_M1[src] * mu_W1[src];
            wgt[NET_SZ + dst] = al_M1[src] * al_W1[src];
        }
        for (int e = tid; e < DD * HH; e += 256) {       // W2: K=64, N=16
            int k = e & 63, n = e >> 6;
            int src = flow * DD * HH + n * HH + k;       // M2[n][k]*W2[n][k]
            int dst = OFF_W2P + (k >> 1) * W2_S + n * 2 + (k & 1);
            wgt[dst]          = mu_M2[src] * mu_W2[src];
            wgt[NET_SZ + dst] = al_M2[src] * al_W2[src];
        }

        // biases -> registers (per-lane, straight from global)
        float bias0[2][4], bias1[2][4], bias2[2];
#pragma unroll
        for (int nt = 0; nt < 4; ++nt) {
            bias0[0][nt] = mu_b0[flow * HH + nt * 16 + l15];
            bias0[1][nt] = al_b0[flow * HH + nt * 16 + l15];
            bias1[0][nt] = mu_b1[flow * HH + nt * 16 + l15];
            bias1[1][nt] = al_b1[flow * HH + nt * 16 + l15];
        }
        bias2[0] = mu_b2[flow * DD + l15];
        bias2[1] = al_b2[flow * DD + l15];

        // stage x into temp (actB, wave-private) for permutation
        for (int idx = lane; idx < 256; idx += 32)
            actB[idx] = xw[(idx >> 4) * XW_S + (idx & 15)];
        __syncthreads();  // weights visible to all waves (also orders temp copy)
        // apply permutation: x_new[:, d] = x_old[:, perm[d]]
        for (int idx = lane; idx < 256; idx += 32) {
            int m = idx >> 4, d = idx & 15;
            xw[m * XW_S + d] = actB[(m << 4) + perms[flow * DD + d]];
        }
        __builtin_amdgcn_wave_barrier();  // wave-private ordering only

        // ---- 16 sequential autoregressive steps ----
        for (int j = 0; j < DD; ++j) {
            // x A-fragments (shared by both nets' layer 0), single b64 each
            v2f xa[4];
#pragma unroll
            for (int kk4 = 0; kk4 < 4; ++kk4)
                xa[kk4] = *(const v2f*)(xw + l15 * XW_S + kk4 * 4 + hl * 2);

            float muv[8];
            for (int net = 0; net < 2; ++net) {
                const float* wg  = wgt + net * NET_SZ;
                const float* w0p = wg + OFF_W0P;
                const float* w1p = wg + OFF_W1P;
                const float* w2p = wg + OFF_W2P;

                // ---- layer 0: x(16x16) @ W0t(16x64) + b0 -> tanh -> actA ----
                {
                    v8f c[4] = {};
#pragma unroll
                    for (int kk4 = 0; kk4 < 4; ++kk4) {
                        const int prow = kk4 * 2 + hl;       // K-pair row
#pragma unroll
                        for (int nt = 0; nt < 4; ++nt) {
                            v2f bv = *(const v2f*)(w0p + prow * W01_S + (nt * 16 + l15) * 2);
                            c[nt] = __builtin_amdgcn_wmma_f32_16x16x4_f32(
                                        false, xa[kk4], false, bv, (short)0, c[nt], false, false);
                        }
                    }
#pragma unroll
                    for (int nt = 0; nt < 4; ++nt) {
                        const int n = nt * 16 + l15;
#pragma unroll
                        for (int r = 0; r < 8; ++r)
                            actA[(r + hl * 8) * ACT_S + n] = fast_tanh(c[nt][r] + bias0[net][nt]);
                    }
                }
                __builtin_amdgcn_wave_barrier();

                // ---- layer 1: actA(16x64) @ W1t(64x64) + b1 -> tanh -> actB ----
                {
                    v8f c[4] = {};
#pragma unroll
                    for (int kk = 0; kk < HH; kk += 4) {
                        v2f av = *(const v2f*)(actA + l15 * ACT_S + kk + hl * 2);
                        const int prow = (kk >> 1) + hl;
#pragma unroll
                        for (int nt = 0; nt < 4; ++nt) {
                            v2f bv = *(const v2f*)(w1p + prow * W01_S + (nt * 16 + l15) * 2);
                            c[nt] = __builtin_amdgcn_wmma_f32_16x16x4_f32(
                                        false, av, false, bv, (short)0, c[nt], false, false);
                        }
                    }
#pragma unroll
                    for (int nt = 0; nt < 4; ++nt) {
                        const int n = nt * 16 + l15;
#pragma unroll
                        for (int r = 0; r < 8; ++r)
                            actB[(r + hl * 8) * ACT_S + n] = fast_tanh(c[nt][r] + bias1[net][nt]);
                    }
                }
                __builtin_amdgcn_wave_barrier();

                // ---- layer 2: actB(16x64) @ W2t(64x16) + b2 -> regs ----
                {
                    v8f c = {};
#pragma unroll
                    for (int kk = 0; kk < HH; kk += 4) {
                        v2f av = *(const v2f*)(actB + l15 * ACT_S + kk + hl * 2);
                        v2f bv = *(const v2f*)(w2p + ((kk >> 1) + hl) * W2_S + l15 * 2);
                        c = __builtin_amdgcn_wmma_f32_16x16x4_f32(
                                false, av, false, bv, (short)0, c, false, false);
                    }
                    if (net == 0) {
#pragma unroll
                        for (int r = 0; r < 8; ++r) muv[r] = c[r] + bias2[0];
                    } else {
                        // sig = 0.1 + 1.9*sigmoid(alpha); update column j of x
                        if (l15 == j) {
#pragma unroll
                            for (int r = 0; r < 8; ++r) {
                                float s = 0.1f + 1.9f * fast_sigmoid(c[r] + bias2[1]);
                                int m = r + hl * 8;
                                xw[m * XW_S + j] = muv[r] + s * xw[m * XW_S + j];
                            }
                        }
                    }
                }
                __builtin_amdgcn_wave_barrier();
            } // net
        } // j
    } // flow

    // write result
    for (int idx = lane; idx < 256; idx += 32) {
        int m = idx >> 4, d = idx & 15;
        out[(size_t)(sampleBase + m) * DD + d] = xw[m * XW_S + d];
    }
}

extern "C" void kernel_launch(void* const* d_in, const int* in_sizes, int n_in,
                              void* d_out, int out_size, void* d_ws, size_t ws_size,
                              hipStream_t stream) {
    const float* mu_M0 = (const float*)d_in[0];
    const float* mu_M1 = (const float*)d_in[1];
    const float* mu_M2 = (const float*)d_in[2];
    const float* mu_W0 = (const float*)d_in[3];
    const float* mu_b0 = (const float*)d_in[4];
    const float* mu_W1 = (const float*)d_in[5];
    const float* mu_b1 = (const float*)d_in[6];
    const float* mu_W2 = (const float*)d_in[7];
    const float* mu_b2 = (const float*)d_in[8];
    const float* al_M0 = (const float*)d_in[9];
    const float* al_M1 = (const float*)d_in[10];
    const float* al_M2 = (const float*)d_in[11];
    const float* al_W0 = (const float*)d_in[12];
    const float* al_b0 = (const float*)d_in[13];
    const float* al_W1 = (const float*)d_in[14];
    const float* al_b1 = (const float*)d_in[15];
    const float* al_W2 = (const float*)d_in[16];
    const float* al_b2 = (const float*)d_in[17];
    const int*   perms = (const int*)d_in[18];
    const float* u     = (const float*)d_in[19];

    const int rows   = in_sizes[19] / DD;              // 65536
    const int blocks = rows / (WAVES_PER_BLOCK * 16);  // 512

    maf_wmma_kernel<<<blocks, 256, LDS_BYTES, stream>>>(
        mu_M0, mu_M1, mu_M2, mu_W0, mu_b0, mu_W1, mu_b1, mu_W2, mu_b2,
        al_M0, al_M1, al_M2, al_W0, al_b0, al_W1, al_b1, al_W2, al_b2,
        perms, u, (float*)d_out);
}